// BidirectionalVSSM_4690104287388
// MI455X (gfx1250) — compile-verified
//
#include <hip/hip_runtime.h>
#include <math.h>

// ---------------- problem constants ----------------
#define BB   8
#define LL   256
#define DD   384
#define EE   768
#define NN   16
#define RR   24
#define MROWS (BB * LL)          // 2048 token rows
#define BLE  (BB * LL * EE)      // 1,572,864
#define XDBL_LD 64               // padded from 56

typedef float v2f __attribute__((ext_vector_type(2)));
typedef float v8f __attribute__((ext_vector_type(8)));

// Async bulk copy global->LDS (16B per lane per op) + wait helpers.
// lds_byte_off: low 32 bits of a generic pointer to __shared__ == LDS offset.
#define ASYNC_LDS_B128(lds_byte_off, gptr)                                   \
    asm volatile("global_load_async_to_lds_b128 %0, %1, off"                 \
                 :: "v"(lds_byte_off), "v"(gptr) : "memory")
#define WAIT_ASYNC0() asm volatile("s_wait_asynccnt 0x0" ::: "memory")

// ---------------------------------------------------------------------------
// fp32 WMMA fragment layout (V_WMMA_F32_16X16X4_F32):
// A: 16x4, lane holds row M=lane%16, K-pair (lane/16)*2..+1   (2 VGPRs)
// B: 4x16, lane holds col N=lane%16, same K-pair              (2 VGPRs)
// C/D: 16x16, VGPR r holds M = r + 8*(lane/16), N = lane%16   (8 VGPRs)
// ---------------------------------------------------------------------------
__device__ __forceinline__ v8f wmma_f32_rowmajorA_wT(
    const float* __restrict__ A, int lda,
    const float* __restrict__ Bw, int ldb,
    int m0, int n0, int K, v8f acc)
{
    const int lane  = threadIdx.x & 31;
    const int khalf = (lane >> 4) * 2;
    const float* ap = A  + (size_t)(m0 + (lane & 15)) * lda + khalf;
    const float* bp = Bw + (size_t)(n0 + (lane & 15)) * ldb + khalf;
#pragma unroll 4
    for (int k = 0; k < K; k += 4) {
        v2f a; a.x = ap[k]; a.y = ap[k + 1];
        v2f b; b.x = bp[k]; b.y = bp[k + 1];
        acc = __builtin_amdgcn_wmma_f32_16x16x4_f32(false, a, false, b,
                                                    (short)0, acc, false, false);
    }
    return acc;
}

// ---------------------------------------------------------------------------
// Kernel 1: xz = x @ W_in^T, split-stored branchlessly: x_inner at xz[..],
// z at xz[.. + BLE] (workspace guarantees z == x_inner + BLE).
// M=2048, N=1536, K=384. block=256 (8 waves), grid=(12, 128).
// A tile (16x384, contiguous 24KB) staged via async-to-LDS; shared by 8 waves.
// ---------------------------------------------------------------------------
__global__ void gemm_in_kernel(const float* __restrict__ x,
                               const float* __restrict__ W_in,
                               float* __restrict__ xz)
{
    __shared__ float sA[16 * DD];                       // 24 KB
    const int wave = threadIdx.x >> 5;
    const int lane = threadIdx.x & 31;
    const int n0 = (blockIdx.x * 8 + wave) * 16;
    const int m0 = blockIdx.y * 16;

    {   // contiguous 6144-float copy: 6 x b128 per thread
        const float* gA = x + (size_t)m0 * DD;
        const unsigned lA = (unsigned)(uintptr_t)&sA[0];
#pragma unroll
        for (int j = 0; j < 6; ++j) {
            const int off = (threadIdx.x + 256 * j) * 4;     // float index
            ASYNC_LDS_B128(lA + (unsigned)(off * 4), gA + off);
        }
        WAIT_ASYNC0();
    }
    __syncthreads();

    const int khalf = (lane >> 4) * 2;
    const float* ap = &sA[(lane & 15) * DD + khalf];         // LDS reads
    const float* bp = W_in + (size_t)(n0 + (lane & 15)) * DD + khalf;

    v8f acc = {};
#pragma unroll 4
    for (int k = 0; k < DD; k += 4) {
        v2f a; a.x = ap[k]; a.y = ap[k + 1];
        v2f b; b.x = bp[k]; b.y = bp[k + 1];
        acc = __builtin_amdgcn_wmma_f32_16x16x4_f32(false, a, false, b,
                                                    (short)0, acc, false, false);
    }

    const int col     = n0 + (lane & 15);                    // 0..1535
    const size_t shft = (col >= EE) ? (size_t)(BLE - EE) : 0; // branchless split
    const int rbase   = m0 + (lane >> 4) * 8;
#pragma unroll
    for (int r = 0; r < 8; ++r)
        xz[(size_t)(rbase + r) * EE + col + shft] = acc[r];
}

// ---------------------------------------------------------------------------
// Kernel 2: x_dbl = x_inner @ W_x^T. M=2048, N=56 (pad 64), K=768.
// block=128 (4 waves cover 64 cols), grid=(1, 128).
// ---------------------------------------------------------------------------
__global__ void gemm_xdbl_kernel(const float* __restrict__ x_inner,
                                 const float* __restrict__ W_x,
                                 float* __restrict__ x_dbl)
{
    const int wave = threadIdx.x >> 5;
    const int lane = threadIdx.x & 31;
    const int n0 = wave * 16;
    const int m0 = blockIdx.y * 16;

    const int lanecol = n0 + (lane & 15);
    const int nsafe   = lanecol < (RR + 2 * NN) ? lanecol : (RR + 2 * NN - 1);
    const int khalf   = (lane >> 4) * 2;
    const float* ap = x_inner + (size_t)(m0 + (lane & 15)) * EE + khalf;
    const float* bp = W_x + (size_t)nsafe * EE + khalf;

    v8f acc = {};
#pragma unroll 4
    for (int k = 0; k < EE; k += 4) {
        v2f a; a.x = ap[k]; a.y = ap[k + 1];
        v2f b; b.x = bp[k]; b.y = bp[k + 1];
        acc = __builtin_amdgcn_wmma_f32_16x16x4_f32(false, a, false, b,
                                                    (short)0, acc, false, false);
    }

    const int rbase = m0 + (lane >> 4) * 8;
#pragma unroll
    for (int r = 0; r < 8; ++r)
        x_dbl[(size_t)(rbase + r) * XDBL_LD + lanecol] = acc[r];
}

// ---------------------------------------------------------------------------
// Kernel 3: dt = softplus( x_dbl[:, :24] @ W_dt^T + b_dt ).
// M=2048, N=768, K=24. block=256, grid=(6, 128).
// ---------------------------------------------------------------------------
__global__ void gemm_dt_kernel(const float* __restrict__ x_dbl,
                               const float* __restrict__ W_dt,
                               const float* __restrict__ b_dt,
                               float* __restrict__ dt)
{
    const int wave = threadIdx.x >> 5;
    const int lane = threadIdx.x & 31;
    const int n0 = blockIdx.x * 128 + wave * 16;
    const int m0 = blockIdx.y * 16;

    v8f acc = {};
    acc = wmma_f32_rowmajorA_wT(x_dbl, XDBL_LD, W_dt, RR, m0, n0, RR, acc);

    const int col   = n0 + (lane & 15);
    const int rbase = m0 + (lane >> 4) * 8;
    const float bias = b_dt[col];
#pragma unroll
    for (int r = 0; r < 8; ++r) {
        float v = acc[r] + bias;
        v = (v > 20.0f) ? v : log1pf(__expf(v));   // softplus
        dt[(size_t)(rbase + r) * EE + col] = v;
    }
}

// ---------------------------------------------------------------------------
// Kernel 4: four directional scans. grid=(3 e-blocks, 8 batches, 4 dirs),
// block=256; thread owns channel e and its 16 hidden states in registers.
// B/C coefficients (32 KB) staged via async-to-LDS; next-token dt/x rows
// prefetched each step (serial-dependency latency hiding).
// ---------------------------------------------------------------------------
__global__ void scan_kernel(const float* __restrict__ x_inner,
                            const float* __restrict__ x_dbl,
                            const float* __restrict__ dt,
                            const float* __restrict__ A_log,
                            float* __restrict__ y_dir)   // [4][B][L][E]
{
    const int b   = blockIdx.y;
    const int dir = blockIdx.z;
    const int e   = blockIdx.x * 256 + threadIdx.x;

    __shared__ float sB[LL * NN];   // 16 KB
    __shared__ float sC[LL * NN];   // 16 KB
    {   // thread p stages row p's 16 B-coeffs + 16 C-coeffs (4+4 x b128)
        const int p = threadIdx.x;
        const float* gB = x_dbl + (size_t)(b * LL + p) * XDBL_LD + RR;
        const float* gC = gB + NN;
        const unsigned lB = (unsigned)(uintptr_t)&sB[p * NN];
        const unsigned lC = (unsigned)(uintptr_t)&sC[p * NN];
#pragma unroll
        for (int j = 0; j < 4; ++j) {
            ASYNC_LDS_B128(lB + 16u * j, gB + 4 * j);
            ASYNC_LDS_B128(lC + 16u * j, gC + 4 * j);
        }
        WAIT_ASYNC0();
    }
    __syncthreads();

    float An[NN];
#pragma unroll
    for (int n = 0; n < NN; ++n) An[n] = -__expf(A_log[e * NN + n]);

    float h[NN];
#pragma unroll
    for (int n = 0; n < NN; ++n) h[n] = 0.0f;

    float* yout = y_dir + (size_t)dir * BLE;

    for (int t = 0; t < LL; ++t) {
        const int tt = (dir & 1) ? (LL - 1 - t) : t;
        const int p  = (dir < 2) ? tt : ((tt & 15) * 16 + (tt >> 4));
        const size_t idx = (size_t)(b * LL + p) * EE + e;

        if (t + 1 < LL) {   // prefetch next step's rows (global_prefetch_b8)
            const int t2  = t + 1;
            const int tt2 = (dir & 1) ? (LL - 1 - t2) : t2;
            const int p2  = (dir < 2) ? tt2 : ((tt2 & 15) * 16 + (tt2 >> 4));
            const size_t idx2 = (size_t)(b * LL + p2) * EE + e;
            __builtin_prefetch(dt + idx2, 0, 1);
            __builtin_prefetch(x_inner + idx2, 0, 1);
        }

        const float dtv = dt[idx];
        const float xv  = x_inner[idx];
        const float dbx = dtv * xv;
        const float* Bp = &sB[p * NN];
        const float* Cp = &sC[p * NN];

        float y = 0.0f;
#pragma unroll
        for (int n = 0; n < NN; ++n) {
            const float a = __expf(dtv * An[n]);          // TRANS pipe
            h[n] = fmaf(a, h[n], dbx * Bp[n]);
            y    = fmaf(h[n], Cp[n], y);
        }
        yout[idx] = y;
    }
}

// ---------------------------------------------------------------------------
// Kernel 5: y = 0.25*(sum of 4 dirs); y = y*silu(z) + x_inner*D_param.
// Writes y_final in-place over y_dir[0] (elementwise, safe).
// ---------------------------------------------------------------------------
__global__ void combine_kernel(const float* __restrict__ x_inner,
                               const float* __restrict__ z,
                               const float* __restrict__ D_param,
                               float* __restrict__ y_dir)
{
    const size_t i = (size_t)blockIdx.x * 256 + threadIdx.x;
    const int e = (int)(i % EE);
    const float y = 0.25f * (y_dir[i] + y_dir[i + (size_t)BLE] +
                             y_dir[i + 2 * (size_t)BLE] + y_dir[i + 3 * (size_t)BLE]);
    const float zv  = z[i];
    const float sil = zv / (1.0f + __expf(-zv));
    y_dir[i] = fmaf(y, sil, x_inner[i] * D_param[e]);
}

// ---------------------------------------------------------------------------
// Kernel 6: out_pre = y_final @ W_out^T + x (residual).
// M=2048, N=384, K=768. block=256, grid=(3, 128).
// A tile (16x768, contiguous 48KB) staged via async-to-LDS; shared by 8 waves.
// ---------------------------------------------------------------------------
__global__ void gemm_out_kernel(const float* __restrict__ y_final,
                                const float* __restrict__ W_out,
                                const float* __restrict__ x_resid,
                                float* __restrict__ out_pre)
{
    __shared__ float sA[16 * EE];                       // 48 KB
    const int wave = threadIdx.x >> 5;
    const int lane = threadIdx.x & 31;
    const int n0 = blockIdx.x * 128 + wave * 16;
    const int m0 = blockIdx.y * 16;

    {   // contiguous 12288-float copy: 12 x b128 per thread
        const float* gA = y_final + (size_t)m0 * EE;
        const unsigned lA = (unsigned)(uintptr_t)&sA[0];
#pragma unroll
        for (int j = 0; j < 12; ++j) {
            const int off = (threadIdx.x + 256 * j) * 4;
            ASYNC_LDS_B128(lA + (unsigned)(off * 4), gA + off);
        }
        WAIT_ASYNC0();
    }
    __syncthreads();

    const int khalf = (lane >> 4) * 2;
    const float* ap = &sA[(lane & 15) * EE + khalf];
    const float* bp = W_out + (size_t)(n0 + (lane & 15)) * EE + khalf;

    v8f acc = {};
#pragma unroll 4
    for (int k = 0; k < EE; k += 4) {
        v2f a; a.x = ap[k]; a.y = ap[k + 1];
        v2f b; b.x = bp[k]; b.y = bp[k + 1];
        acc = __builtin_amdgcn_wmma_f32_16x16x4_f32(false, a, false, b,
                                                    (short)0, acc, false, false);
    }

    const int col   = n0 + (lane & 15);
    const int rbase = m0 + (lane >> 4) * 8;
#pragma unroll
    for (int r = 0; r < 8; ++r) {
        const size_t o = (size_t)(rbase + r) * DD + col;
        out_pre[o] = acc[r] + x_resid[o];
    }
}

// ---------------------------------------------------------------------------
// Kernel 7: LayerNorm over D=384 per row. block=128, grid=2048.
// ---------------------------------------------------------------------------
__global__ void ln_kernel(const float* __restrict__ out_pre,
                          const float* __restrict__ gamma,
                          const float* __restrict__ beta,
                          float* __restrict__ out)
{
    const int row = blockIdx.x;
    const int tid = threadIdx.x;
    const float* rp = out_pre + (size_t)row * DD;

    float v0 = rp[tid], v1 = rp[tid + 128], v2 = rp[tid + 256];
    float s  = v0 + v1 + v2;
    float s2 = v0 * v0 + v1 * v1 + v2 * v2;

#pragma unroll
    for (int off = 16; off > 0; off >>= 1) {        // wave32 reduction
        s  += __shfl_xor(s,  off, 32);
        s2 += __shfl_xor(s2, off, 32);
    }
    __shared__ float red[8];
    const int wv = tid >> 5;
    if ((tid & 31) == 0) { red[wv] = s; red[4 + wv] = s2; }
    __syncthreads();
    const float ts  = red[0] + red[1] + red[2] + red[3];
    const float ts2 = red[4] + red[5] + red[6] + red[7];

    const float mu  = ts * (1.0f / DD);
    const float var = ts2 * (1.0f / DD) - mu * mu;
    const float inv = rsqrtf(var + 1e-5f);

    float* op = out + (size_t)row * DD;
#pragma unroll
    for (int j = 0; j < 3; ++j) {
        const int c = tid + j * 128;
        op[c] = (rp[c] - mu) * inv * gamma[c] + beta[c];
    }
}

// ---------------------------------------------------------------------------
// Launch. Workspace layout (floats):
//   x_inner [BLE] | z [BLE] | x_dbl [2048*64] | dt [BLE] | y_dir [4*BLE]
//   (z MUST be x_inner + BLE: gemm_in relies on it for the branchless split)
//   y_final aliases y_dir[0]; out_pre aliases dt (dt fully consumed by then).
// Total ~44.6 MB.
// ---------------------------------------------------------------------------
extern "C" void kernel_launch(void* const* d_in, const int* in_sizes, int n_in,
                              void* d_out, int out_size, void* d_ws, size_t ws_size,
                              hipStream_t stream)
{
    (void)in_sizes; (void)n_in; (void)out_size; (void)ws_size;
    const float* x       = (const float*)d_in[0];
    const float* W_in    = (const float*)d_in[1];
    const float* A_log   = (const float*)d_in[2];
    const float* W_x     = (const float*)d_in[3];
    const float* W_dt    = (const float*)d_in[4];
    const float* b_dt    = (const float*)d_in[5];
    const float* D_param = (const float*)d_in[6];
    const float* W_out   = (const float*)d_in[7];
    const float* gamma   = (const float*)d_in[8];
    const float* beta    = (const float*)d_in[9];
    float* out = (float*)d_out;

    float* ws      = (float*)d_ws;
    float* x_inner = ws;                                  // == xz base
    float* z       = x_inner + BLE;                       // contiguity required
    float* x_dbl   = z + BLE;
    float* dt      = x_dbl + (size_t)MROWS * XDBL_LD;
    float* y_dir   = dt + BLE;
    float* y_final = y_dir;        // alias dir0 after combine
    float* out_pre = dt;           // reuse dt region

    gemm_in_kernel<<<dim3(12, MROWS / 16), 256, 0, stream>>>(x, W_in, x_inner);
    gemm_xdbl_kernel<<<dim3(1, MROWS / 16), 128, 0, stream>>>(x_inner, W_x, x_dbl);
    gemm_dt_kernel<<<dim3(6, MROWS / 16), 256, 0, stream>>>(x_dbl, W_dt, b_dt, dt);
    scan_kernel<<<dim3(3, BB, 4), 256, 0, stream>>>(x_inner, x_dbl, dt, A_log, y_dir);
    combine_kernel<<<dim3(BLE / 256), 256, 0, stream>>>(x_inner, z, D_param, y_dir);
    gemm_out_kernel<<<dim3(3, MROWS / 16), 256, 0, stream>>>(y_final, W_out, x, out_pre);
    ln_kernel<<<dim3(MROWS), 128, 0, stream>>>(out_pre, gamma, beta, out);
}